// SSIM_83863531422305
// MI455X (gfx1250) — compile-verified
//
#include <hip/hip_runtime.h>

// SSIM loss, fused single pass over 16x3x512x512 fp32 images on MI455X.
// All five depthwise 11x11 Gaussian convs are done as banded fp32 matmuls on
// the matrix pipes via V_WMMA_F32_16X16X4_F32 (the only fp32 WMMA on CDNA5),
// keeping bit-compatible fp32 math with the reference while offloading the
// ~2.8 GFLOP of conv work from the VALU.

typedef __attribute__((ext_vector_type(2))) float v2f;
typedef __attribute__((ext_vector_type(8))) float v8f;

#define H_IMG 512
#define W_IMG 512
#define N_IMG 48            // 16 batch * 3 channels
#define TILE_H 16
#define TILE_W 64
#define IN_ROWS 28          // 26 valid halo rows + 2 zero rows (K padded 26->28)
#define IN_COLS 80          // 64 + 2*5 halo + pad read room for K-chunk 6
#define LDS_PITCH 84        // 84 mod 64 spreads banks: conflict-free T gather

// normalized 11-tap Gaussian, sigma=1.5 (matches reference _make_window)
__device__ __forceinline__ float gtab(int i) {
    const float g[11] = {
        0.00102838f, 0.00759874f, 0.03600077f, 0.10936069f, 0.21300553f,
        0.26601172f,
        0.21300553f, 0.10936069f, 0.03600077f, 0.00759874f, 0.00102838f };
    return (i >= 0 && i <= 10) ? g[i] : 0.0f;
}

__device__ __forceinline__ v8f wmma_f32(v2f a, v2f b, v8f c) {
    // D = A(16x4 f32) * B(4x16 f32) + C(16x16 f32)
    return __builtin_amdgcn_wmma_f32_16x16x4_f32(
        /*neg_a=*/false, a, /*neg_b=*/false, b,
        /*c_mod=*/(short)0, c, /*reuse_a=*/false, /*reuse_b=*/false);
}

__global__ __launch_bounds__(32)
void ssim_tile_kernel(const float* __restrict__ img1,
                      const float* __restrict__ img2,
                      double* __restrict__ acc_out) {
    __shared__ float sIn1[IN_ROWS * LDS_PITCH];
    __shared__ float sIn2[IN_ROWS * LDS_PITCH];
    __shared__ float sT[5][TILE_H * LDS_PITCH];   // vertically-filtered tiles

    const int lane = threadIdx.x;   // wave32: one wave per block
    const int n    = lane & 15;     // matrix row/col index within half-wave
    const int hh   = lane >> 4;     // half-wave select (K offset +2 / M offset +8)

    const int x0 = blockIdx.x * TILE_W;
    const int y0 = blockIdx.y * TILE_H;
    const size_t base = (size_t)blockIdx.z * (size_t)(H_IMG * W_IMG);

    // ---- load halo tiles (SAME zero padding; rows 26/27 zeroed for K padding)
    for (int idx = lane; idx < IN_ROWS * IN_COLS; idx += 32) {
        const int r  = idx / IN_COLS;
        const int c  = idx - r * IN_COLS;
        const int gy = y0 - 5 + r;
        const int gx = x0 - 5 + c;
        float a = 0.0f, b = 0.0f;
        if (r < 26 && gy >= 0 && gy < H_IMG && gx >= 0 && gx < W_IMG) {
            const size_t off = base + (size_t)gy * W_IMG + gx;
            a = img1[off];
            b = img2[off];
        }
        sIn1[r * LDS_PITCH + c] = a;
        sIn2[r * LDS_PITCH + c] = b;
    }
    // single wave per workgroup: LDS ops from one wave are in-order -> no barrier

    // Banded Gaussian chunks. Gv[m][j]=g[j-m] (vertical A) and Gh[j][n]=g[j-n]
    // (horizontal B) have IDENTICAL per-lane register images under the CDNA5
    // A-/B-matrix layouts, so w[] serves both passes.
    v2f w[7];
#pragma unroll
    for (int t = 0; t < 7; ++t) {
        const int k = 4 * t + 2 * hh;
        w[t].x = gtab(k - n);
        w[t].y = gtab(k + 1 - n);
    }

    // ---- phase 1: vertical pass  T[ch] = Gv(16x28) * In_ch(28x16) per col group
    for (int cg = 0; cg < 5; ++cg) {
        v8f a0v = {}; v8f a1v = {}; v8f a2v = {}; v8f a3v = {}; v8f a4v = {};
        const int c = cg * 16 + n;
#pragma unroll
        for (int t = 0; t < 7; ++t) {
            const int k = 4 * t + 2 * hh;      // B rows k, k+1 for this half-wave
            const float p0 = sIn1[k * LDS_PITCH + c];
            const float p1 = sIn1[(k + 1) * LDS_PITCH + c];
            const float q0 = sIn2[k * LDS_PITCH + c];
            const float q1 = sIn2[(k + 1) * LDS_PITCH + c];
            const v2f b0 = { p0, p1 };              // img1
            const v2f b1 = { q0, q1 };              // img2
            const v2f b2 = { p0 * p0, p1 * p1 };    // img1^2
            const v2f b3 = { q0 * q0, q1 * q1 };    // img2^2
            const v2f b4 = { p0 * q0, p1 * q1 };    // img1*img2
            a0v = wmma_f32(w[t], b0, a0v);
            a1v = wmma_f32(w[t], b1, a1v);
            a2v = wmma_f32(w[t], b2, a2v);
            a3v = wmma_f32(w[t], b3, a3v);
            a4v = wmma_f32(w[t], b4, a4v);
        }
#pragma unroll
        for (int i = 0; i < 8; ++i) {
            // D layout: VGPR i -> row i (lanes 0-15) / row i+8 (lanes 16-31)
            const int row = i + 8 * hh;
            const int col = cg * 16 + n;
            sT[0][row * LDS_PITCH + col] = a0v[i];
            sT[1][row * LDS_PITCH + col] = a1v[i];
            sT[2][row * LDS_PITCH + col] = a2v[i];
            sT[3][row * LDS_PITCH + col] = a3v[i];
            sT[4][row * LDS_PITCH + col] = a4v[i];
        }
    }

    // ---- phase 2: horizontal pass  Out = T(16x28 slice) * Gh(28x16) + SSIM map
    const float C1 = 1.0e-4f;   // (0.01*L)^2
    const float C2 = 9.0e-4f;   // (0.03*L)^2
    float lsum = 0.0f;
    for (int gq = 0; gq < 4; ++gq) {
        v8f r0 = {}; v8f r1 = {}; v8f r2 = {}; v8f r3 = {}; v8f r4 = {};
        const int kb = gq * 16;
#pragma unroll
        for (int t = 0; t < 7; ++t) {
            const int k = kb + 4 * t + 2 * hh;   // A cols k, k+1 for this half-wave
            const int rowoff = n * LDS_PITCH + k;
            const v2f a0 = { sT[0][rowoff], sT[0][rowoff + 1] };
            const v2f a1 = { sT[1][rowoff], sT[1][rowoff + 1] };
            const v2f a2 = { sT[2][rowoff], sT[2][rowoff + 1] };
            const v2f a3 = { sT[3][rowoff], sT[3][rowoff + 1] };
            const v2f a4 = { sT[4][rowoff], sT[4][rowoff + 1] };
            r0 = wmma_f32(a0, w[t], r0);
            r1 = wmma_f32(a1, w[t], r1);
            r2 = wmma_f32(a2, w[t], r2);
            r3 = wmma_f32(a3, w[t], r3);
            r4 = wmma_f32(a4, w[t], r4);
        }
#pragma unroll
        for (int i = 0; i < 8; ++i) {
            const float mu1   = r0[i];
            const float mu2   = r1[i];
            const float mu1sq = mu1 * mu1;
            const float mu2sq = mu2 * mu2;
            const float mu12  = mu1 * mu2;
            const float s1  = r2[i] - mu1sq;
            const float s2  = r3[i] - mu2sq;
            const float s12 = r4[i] - mu12;
            const float num = (2.0f * mu12 + C1) * (2.0f * s12 + C2);
            const float den = (mu1sq + mu2sq + C1) * (s1 + s2 + C2);
            lsum += num / den;
        }
    }

    // ---- wave32 reduction + one fp64 atomic per tile
    for (int off = 16; off > 0; off >>= 1)
        lsum += __shfl_down(lsum, off, 32);
    if (lane == 0)
        atomicAdd(acc_out, (double)lsum);
}

__global__ void ssim_zero_kernel(double* __restrict__ acc) {
    if (threadIdx.x == 0 && blockIdx.x == 0) acc[0] = 0.0;
}

__global__ void ssim_finalize_kernel(const double* __restrict__ acc,
                                     float* __restrict__ out) {
    if (threadIdx.x == 0 && blockIdx.x == 0) {
        const double n = (double)N_IMG * (double)H_IMG * (double)W_IMG;
        out[0] = 1.0f - (float)(acc[0] / n);
    }
}

extern "C" void kernel_launch(void* const* d_in, const int* in_sizes, int n_in,
                              void* d_out, int out_size, void* d_ws, size_t ws_size,
                              hipStream_t stream) {
    const float* img1 = (const float*)d_in[0];
    const float* img2 = (const float*)d_in[1];
    double* acc = (double*)d_ws;
    float* out  = (float*)d_out;

    hipLaunchKernelGGL(ssim_zero_kernel, dim3(1), dim3(1), 0, stream, acc);

    dim3 grid(W_IMG / TILE_W, H_IMG / TILE_H, N_IMG);   // 8 x 32 x 48 tiles
    hipLaunchKernelGGL(ssim_tile_kernel, grid, dim3(32), 0, stream,
                       img1, img2, acc);

    hipLaunchKernelGGL(ssim_finalize_kernel, dim3(1), dim3(1), 0, stream,
                       acc, out);
}